// CartesianMACE_16810501997278
// MI455X (gfx1250) — compile-verified
//
#include <hip/hip_runtime.h>
#include <math.h>

typedef float v2f __attribute__((ext_vector_type(2)));
typedef float v8f __attribute__((ext_vector_type(8)));
typedef unsigned int v4u __attribute__((ext_vector_type(4)));
typedef int v8i __attribute__((ext_vector_type(8)));
typedef int v4i __attribute__((ext_vector_type(4)));

constexpr int N_NODES = 10000;
constexpr int N_CH    = 32;
constexpr int N_EDGES = 160000;
constexpr int HCOLS   = 13;            // packed cartesian cols (output layout)
constexpr int HP      = 16;            // padded cols for internal buffers
constexpr int HSTR    = N_CH * HP;     // 512 floats per node (internal)
constexpr int KTOT    = 96;            // 3 feature ranks * 32 ch
constexpr int NTOT    = 96;            // 3 basis  ranks * 32 ch
constexpr int SSTR    = 97;            // padded S-tile row stride
constexpr int WSTR    = 65;            // node weight row stride ([Cw|Mw] + pad)
constexpr int XSTR    = 17;            // node X-tile row stride

__device__ __forceinline__ v8f wmma_f32(v2f a, v2f b, v8f c) {
  // V_WMMA_F32_16X16X4_F32: (neg_a, A, neg_b, B, c_mod, C, reuse_a, reuse_b)
  return __builtin_amdgcn_wmma_f32_16x16x4_f32(false, a, false, b, (short)0, c,
                                               false, false);
}

#define HAS_TDM __has_builtin(__builtin_amdgcn_tensor_load_to_lds)

// TDM: load a 32x32 f32 tile (contiguous rows of 32 dwords in memory) into LDS,
// inserting 33 pad dwords after every 32 stored dwords -> LDS row stride 65.
__device__ __forceinline__ void tdm_load_32x32_pad65(const float* gsrc,
                                                     unsigned lds_byte) {
#if HAS_TDM
  unsigned long long ga = (unsigned long long)(uintptr_t)gsrc;
  unsigned lds_u = (unsigned)__builtin_amdgcn_readfirstlane((int)lds_byte);
  unsigned galo  = (unsigned)__builtin_amdgcn_readfirstlane((int)(unsigned)ga);
  unsigned gahi  = (unsigned)__builtin_amdgcn_readfirstlane((int)(unsigned)(ga >> 32));
  v4u g0;
  g0[0] = 1u;                                       // count = 1 valid descriptor
  g0[1] = lds_u;                                    // lds_addr (bytes)
  g0[2] = galo;                                     // global_addr[31:0]
  g0[3] = (gahi & 0x1FFFFFFu) | (2u << 30);         // global_addr[56:32] | type=2
  v8i g1;
  g1[0] = 0x41120000;  // data_size=4B, pad_enable, pad_interval=32dw, pad_amt=33dw
  g1[1] = 32 << 16;    // tensor_dim0 = 32 (bits 63:48)
  g1[2] = 32 << 16;    // tensor_dim1 = 32 (bits 95:80)
  g1[3] = 32 << 16;    // tile_dim0   = 32 (bits 127:112)
  g1[4] = 32;          // tile_dim1   = 32
  g1[5] = 32;          // tensor_dim0_stride = 32
  g1[6] = 0;
  g1[7] = 0;
  v4i z4 = {0, 0, 0, 0};
  v8i z8 = {0, 0, 0, 0, 0, 0, 0, 0};
  // 6-arg toolchain variant: (g0, g1, g2, g3, g4, cpol)
  __builtin_amdgcn_tensor_load_to_lds(g0, g1, z4, z4, z8, 0);
#else
  (void)gsrc; (void)lds_byte;
#endif
}

// ---------------------------------------------------------------- init / zero
__global__ void init_h_kernel(const float* __restrict__ emb_w,
                              const int* __restrict__ atoms,
                              float* __restrict__ h) {
  int i = blockIdx.x * blockDim.x + threadIdx.x;
  if (i >= N_NODES * HSTR) return;
  int col = i % HP;
  int ch  = (i / HP) % N_CH;
  int n   = i / HSTR;
  h[i] = (col == 0) ? emb_w[atoms[n] * N_CH + ch] : 0.0f;
}

__global__ void zero_kernel(float* __restrict__ p, int n) {
  int i = blockIdx.x * blockDim.x + threadIdx.x;
  if (i < n) p[i] = 0.0f;
}

// ------------------------------------------------------------------ edge pass
#define EW_WAVES 2
__global__ __launch_bounds__(EW_WAVES * 32)
void edge_kernel(const float* __restrict__ pos,
                 const int* __restrict__ edge_index,
                 const float* __restrict__ W_ab,
                 const float* __restrict__ h,
                 float* __restrict__ A, int layer) {
  __shared__ float Wlds[KTOT * NTOT];               // 36.9 KB, WG-shared
  __shared__ float Slds[EW_WAVES][16 * SSTR];       // per-wave S tile (padded)
  __shared__ float Ulds[EW_WAVES][16 * HP];         // per-wave u-powers
  __shared__ int   Dlds[EW_WAVES][16];              // per-wave dst ids

  const int tid  = threadIdx.x;
  const int wave = tid >> 5;
  const int lane = tid & 31;
  const int eloc = lane & 15;   // edge within tile / N-col within tile
  const int hi   = lane >> 4;   // half-wave: selects K {0,1} vs {2,3}

  // W[k][n] = W_ab[layer, c=n>>5, a=k>>5][j=n&31][kc=k&31]
  for (int i = tid; i < KTOT * NTOT; i += blockDim.x) {
    int k = i / NTOT, n = i % NTOT;
    int a = k >> 5, kc = k & 31, c = n >> 5, j = n & 31;
    Wlds[i] = W_ab[(((layer * 3 + c) * 3 + a) * N_CH + j) * N_CH + kc];
  }
  __syncthreads();

  const int  nTiles = N_EDGES / 16;                 // 10000, exact
  const int  gwave  = blockIdx.x * EW_WAVES + wave;
  const int  nWaves = gridDim.x * EW_WAVES;
  const int* srcArr = edge_index;
  const int* dstArr = edge_index + N_EDGES;

  for (int tile = gwave; tile < nTiles; tile += nWaves) {
    const int e   = tile * 16 + eloc;
    const int src = srcArr[e];
    const int dst = dstArr[e];

    float ux = pos[dst * 3 + 0] - pos[src * 3 + 0];
    float uy = pos[dst * 3 + 1] - pos[src * 3 + 1];
    float uz = pos[dst * 3 + 2] - pos[src * 3 + 2];
    float inv = 1.0f / (sqrtf(ux * ux + uy * uy + uz * uz) + 1e-9f);
    ux *= inv; uy *= inv; uz *= inv;
    float up[HP];
    up[0] = 1.0f; up[1] = ux; up[2] = uy; up[3] = uz;
    up[4]  = ux * ux; up[5]  = ux * uy; up[6]  = ux * uz;
    up[7]  = uy * ux; up[8]  = uy * uy; up[9]  = uy * uz;
    up[10] = uz * ux; up[11] = uz * uy; up[12] = uz * uz;
    up[13] = 0.0f; up[14] = 0.0f; up[15] = 0.0f;
    if (hi == 0) {
      for (int q = 0; q < HP; ++q) Ulds[wave][eloc * HP + q] = up[q];
      Dlds[wave][eloc] = dst;
    }
    // S row of edge eloc: lane covers channels hi*16 .. hi*16+15.
    // Padded 16-float rows (16B aligned) -> 4x b128 gather per channel.
    const float* hp = h + (size_t)src * HSTR;
    for (int cc = 0; cc < 16; ++cc) {
      int ch = hi * 16 + cc;
      const float4* hc4 = (const float4*)(hp + ch * HP);
      float4 r0 = hc4[0], r1 = hc4[1], r2 = hc4[2], r3 = hc4[3];
      float s0 = r0.x;
      float s1 = r0.y * up[1] + r0.z * up[2] + r0.w * up[3];
      float s2 = r1.x * up[4] + r1.y * up[5] + r1.z * up[6] +
                 r1.w * up[7] + r2.x * up[8] + r2.y * up[9] +
                 r2.z * up[10] + r2.w * up[11] + r3.x * up[12];
      Slds[wave][eloc * SSTR + ch]      = s0;
      Slds[wave][eloc * SSTR + 32 + ch] = s1;
      Slds[wave][eloc * SSTR + 64 + ch] = s2;
    }
    __syncthreads();

    v8f D[6] = {};                                  // 6 N-tiles of 16 cols
    const int koff = hi ? 2 : 0;
    for (int kk = 0; kk < KTOT / 4; ++kk) {
      v2f Af;
      Af[0] = Slds[wave][eloc * SSTR + kk * 4 + koff];
      Af[1] = Slds[wave][eloc * SSTR + kk * 4 + koff + 1];
      #pragma unroll
      for (int nt = 0; nt < 6; ++nt) {
        v2f Bf;
        Bf[0] = Wlds[(kk * 4 + koff) * NTOT + nt * 16 + eloc];
        Bf[1] = Wlds[(kk * 4 + koff + 1) * NTOT + nt * 16 + eloc];
        D[nt] = wmma_f32(Af, Bf, D[nt]);
      }
    }

    // scatter: A[dst][j][colbase+d] += D * u^(x c)
    #pragma unroll
    for (int nt = 0; nt < 6; ++nt) {
      const int c = nt >> 1;
      const int j = (nt & 1) * 16 + eloc;           // output channel
      const int colbase = (c == 0) ? 0 : ((c == 1) ? 1 : 4);
      const int wc      = (c == 0) ? 1 : ((c == 1) ? 3 : 9);
      for (int r = 0; r < 8; ++r) {
        int   m    = r + hi * 8;                    // edge within tile
        float val  = D[nt][r];
        int   node = Dlds[wave][m];
        float* ap  = A + (size_t)node * HSTR + j * HP + colbase;
        for (int d = 0; d < wc; ++d)
          atomicAdd(ap + d, val * Ulds[wave][m * HP + colbase + d]);
      }
    }
    __syncthreads();
  }
}

// ------------------------------------------------------------------ node pass
#define NW_WAVES 4
__global__ __launch_bounds__(NW_WAVES * 32)
void node_kernel(const float* __restrict__ ws_w,
                 const float* __restrict__ channel_w,
                 const float* __restrict__ message_w,
                 const float* __restrict__ A,
                 const float* __restrict__ h_in,
                 float* __restrict__ h_out, int layer, int out_cstride) {
  __shared__ float Wl[NW_WAVES][N_CH * WSTR + N_CH]; // per-wave weights, 8.4 KB
  __shared__ float Xl[NW_WAVES][64 * XSTR];          // per-wave B tile, 4.3 KB

  const int tid  = threadIdx.x;
  const int wave = tid >> 5;
  const int lane = tid & 31;
  const int l16  = lane & 15;
  const int hi   = lane >> 4;

  const unsigned wl_byte = (unsigned)(uintptr_t)&Wl[wave][0]; // LDS byte offset
  const int gwave  = blockIdx.x * NW_WAVES + wave;
  const int nWaves = gridDim.x * NW_WAVES;
  const int out_nstride = N_CH * out_cstride;

  for (int n = gwave; n < N_NODES; n += nWaves) {
    float s0 = A[(size_t)n * HSTR + lane * HP];           // channel = lane
    for (int c = 0; c < 3; ++c) {
      const int colbase = (c == 0) ? 0 : ((c == 1) ? 1 : 4);
      const int wc      = (c == 0) ? 1 : ((c == 1) ? 3 : 9);
      const float* wsp = ws_w + (size_t)((layer * 3 + c) * 3) * N_CH;
      float g = wsp[lane] + wsp[N_CH + lane] * s0 + wsp[2 * N_CH + lane] * s0 * s0;

      const float* Cw = channel_w +
          ((size_t)(layer * 3 + c) * N_NODES + n) * (N_CH * N_CH);
      const float* Mw = message_w +
          ((size_t)(layer * 3 + c) * N_NODES + n) * (N_CH * N_CH);

#if HAS_TDM
      // WAR fence: prior iteration's LDS reads must land before TDM overwrite.
      asm volatile("s_wait_dscnt 0x0" ::: "memory");
      tdm_load_32x32_pad65(Cw, wl_byte);                  // rows j*65 + [0..31]
      tdm_load_32x32_pad65(Mw, wl_byte + 32u * 4u);       // rows j*65 + [32..63]
#else
      for (int i = lane; i < N_CH * N_CH; i += 32) {
        int j = i >> 5, k = i & 31;
        Wl[wave][j * WSTR + k]      = Cw[i];
        Wl[wave][j * WSTR + 32 + k] = Mw[i];
      }
#endif
      const float* hrow = h_in + (size_t)n * HSTR + lane * HP + colbase;
      const float* arow = A    + (size_t)n * HSTR + lane * HP + colbase;
      for (int col = 0; col < 16; ++col) {
        float hv = (col < wc) ? hrow[col]     : 0.0f;
        float mv = (col < wc) ? g * arow[col] : 0.0f;
        Xl[wave][lane * XSTR + col]        = hv;
        Xl[wave][(32 + lane) * XSTR + col] = mv;
      }
#if HAS_TDM
      __builtin_amdgcn_s_wait_tensorcnt(0);               // weights resident
#endif
      __syncthreads();

      const int koff = hi ? 2 : 0;
      for (int mt = 0; mt < 2; ++mt) {
        v8f D = {};
        for (int kk = 0; kk < 16; ++kk) {
          v2f Af, Bf;
          int row = mt * 16 + l16;
          Af[0] = Wl[wave][row * WSTR + kk * 4 + koff];
          Af[1] = Wl[wave][row * WSTR + kk * 4 + koff + 1];
          Bf[0] = Xl[wave][(kk * 4 + koff) * XSTR + l16];
          Bf[1] = Xl[wave][(kk * 4 + koff + 1) * XSTR + l16];
          D = wmma_f32(Af, Bf, D);
        }
        if (l16 < wc) {
          for (int r = 0; r < 8; ++r) {
            int j = mt * 16 + hi * 8 + r;
            h_out[(size_t)n * out_nstride + j * out_cstride + colbase + l16] = D[r];
          }
        }
      }
      __syncthreads();
    }
  }
}

// ---------------------------------------------------------------------- launch
extern "C" void kernel_launch(void* const* d_in, const int* in_sizes, int n_in,
                              void* d_out, int out_size, void* d_ws,
                              size_t ws_size, hipStream_t stream) {
  (void)in_sizes; (void)n_in; (void)out_size; (void)ws_size;
  const float* pos        = (const float*)d_in[0];
  const int*   edge_index = (const int*)d_in[1];
  const int*   atoms      = (const int*)d_in[2];
  const float* emb_w      = (const float*)d_in[3];
  const float* W_ab       = (const float*)d_in[4];
  const float* ws_w       = (const float*)d_in[5];
  const float* channel_w  = (const float*)d_in[6];
  const float* message_w  = (const float*)d_in[7];
  float*       out        = (float*)d_out;

  float* ws   = (float*)d_ws;
  float* hA   = ws;                               // N*512 floats (padded)
  float* hB   = ws +     (size_t)N_NODES * HSTR;
  float* Abuf = ws + 2 * (size_t)N_NODES * HSTR;

  const int total = N_NODES * HSTR;
  init_h_kernel<<<(total + 255) / 256, 256, 0, stream>>>(emb_w, atoms, hA);

  float* hcur = hA;
  for (int l = 0; l < 2; ++l) {
    zero_kernel<<<(total + 255) / 256, 256, 0, stream>>>(Abuf, total);
    // 250 blocks * 2 waves = 500 waves -> 20 edge tiles each (uniform)
    edge_kernel<<<250, EW_WAVES * 32, 0, stream>>>(pos, edge_index, W_ab, hcur,
                                                   Abuf, l);
    float* hnext   = (l == 0) ? hB : out;
    int    cstride = (l == 0) ? HP : HCOLS;       // final pass writes packed
    // 250 blocks * 4 waves = 1000 waves -> 10 nodes each (uniform)
    node_kernel<<<250, NW_WAVES * 32, 0, stream>>>(
        ws_w, channel_w, message_w, Abuf, hcur, hnext, l, cstride);
    hcur = hnext;
  }
}